// SentenceMatrixLayer_1580547970927
// MI455X (gfx1250) — compile-verified
//
#include <hip/hip_runtime.h>

// ---- Problem constants (from reference): B=16, S=2048, D=256 ----
#define BSZ 16
#define SEQ 2048
#define DIM 256
#define NROWS (BSZ * SEQ)      // 32768 flattened (b,s) rows
#define P_ASEM 0.6f
#define Q_ASEM 0.4f
#define MASK_FILL 1e-9f

typedef __attribute__((ext_vector_type(2))) float v2f;
typedef __attribute__((ext_vector_type(4))) float f4;
typedef __attribute__((ext_vector_type(8))) float v8f;
typedef __attribute__((ext_vector_type(4))) int   i4;

// -----------------------------------------------------------------------------
// Kernel 1: si/sj projection via V_WMMA_F32_16X16X4_F32.
// One wave32 handles a 16-row tile of X (16 x 256). Accumulate 64 WMMA steps
// (K=4 each). B matrix is (4 x 16) with col 0 = Wi, col 1 = Wj, rest zero.
// A layout  : lane l -> row l%16, VGPR0=K(2*(l>>4)), VGPR1=K+1  (float2 load)
// B layout  : lane l -> col l%16, same K striping as A
// D layout  : VGPR v -> M=v (lanes 0-15) / M=v+8 (lanes 16-31), N = lane%16
//
// To avoid divergent predicated loads in the inner loop, every lane loads a
// W float2 from a pre-selected pointer (wi for col 0, wj for col 1, wi as a
// harmless dummy for cols 2..15 -- W is 2 KB and L1-resident), and the unused
// columns are zeroed with value selects (v_cndmask), not branches.
// -----------------------------------------------------------------------------
__global__ void __launch_bounds__(256)
sisj_wmma_kernel(const float* __restrict__ x,   // (NROWS, DIM)
                 const float* __restrict__ W,   // (2*DIM): [Wi | Wj]
                 float* __restrict__ si,        // (NROWS)
                 float* __restrict__ sj)        // (NROWS)
{
    const int lane  = threadIdx.x & 31;
    const int wave  = threadIdx.x >> 5;
    const int tile  = blockIdx.x * (blockDim.x >> 5) + wave;
    const int row0  = tile * 16;
    const int m     = lane & 15;     // A row-in-tile, and B/D column
    const int khalf = lane >> 4;     // which K pair this lane holds

    const float* xrow = x + (size_t)(row0 + m) * DIM;
    // Pointer select once (v_cndmask on address), no branches in the loop.
    const float* wsel = (m == 1) ? (W + DIM) : W;
    const bool   live = (m < 2);

    v8f acc = {0.f, 0.f, 0.f, 0.f, 0.f, 0.f, 0.f, 0.f};

#pragma unroll 4
    for (int k = 0; k < DIM / 4; ++k) {
        const int kbase = k * 4 + 2 * khalf;
        v2f a  = *(const v2f*)(xrow + kbase);
        v2f wv = *(const v2f*)(wsel + kbase);   // all lanes load (cache hit)
        v2f bb;
        bb.x = live ? wv.x : 0.0f;              // v_cndmask, no exec churn
        bb.y = live ? wv.y : 0.0f;
        // 8 args: (neg_a, A, neg_b, B, c_mod, C, reuse_a, reuse_b)
        acc = __builtin_amdgcn_wmma_f32_16x16x4_f32(
            false, a, false, bb, (short)0, acc, false, false);
    }

    // Column 0 of D = si, column 1 = sj. Lanes 0/16 and 1/17 hold them.
    float* dst = (m == 0) ? si : (m == 1) ? sj : nullptr;
    if (dst) {
        const int base = row0 + khalf * 8;
#pragma unroll
        for (int v = 0; v < 8; ++v) dst[base + v] = acc[v];
    }
}

// -----------------------------------------------------------------------------
// Kernel 2: bandwidth-bound fused pass. One block per (b,i) row, 8 elems/thread.
//  - mask_i == 0: fill whole row with 1e-9, skip adj read entirely (~50% rows).
//  - adj/out are touched once -> non-temporal to keep L2 for sj/mask reuse.
// -----------------------------------------------------------------------------
__global__ void __launch_bounds__(256)
fuse_kernel(const float* __restrict__ adj,   // (NROWS, SEQ)
            const int*   __restrict__ mask,  // (NROWS)
            const float* __restrict__ si,    // (NROWS)
            const float* __restrict__ sj,    // (NROWS)
            const float* __restrict__ bptr,  // (1)
            float*       __restrict__ out)   // (NROWS, SEQ)
{
    const int row  = blockIdx.x;       // b*SEQ + i
    const int bidx = row >> 11;        // / SEQ
    const int j0   = threadIdx.x * 8;

    float* orow = out + (size_t)row * SEQ + j0;

    if (mask[row] == 0) {              // uniform per block -> scalar branch
        f4 fill = {MASK_FILL, MASK_FILL, MASK_FILL, MASK_FILL};
        __builtin_nontemporal_store(fill, (f4*)orow);
        __builtin_nontemporal_store(fill, (f4*)(orow + 4));
        return;
    }

    const float ti = si[row] + bptr[0];

    const float* arow = adj + (size_t)row * SEQ + j0;
    f4 a0 = __builtin_nontemporal_load((const f4*)arow);
    f4 a1 = __builtin_nontemporal_load((const f4*)(arow + 4));

    const float* sjb = sj + (bidx << 11) + j0;       // cached, reused row-wide
    f4 s0 = *(const f4*)sjb;
    f4 s1 = *(const f4*)(sjb + 4);
    const int* mb = mask + (bidx << 11) + j0;
    i4 m0 = *(const i4*)mb;
    i4 m1 = *(const i4*)(mb + 4);

    f4 r0, r1;
#pragma unroll
    for (int e = 0; e < 4; ++e) {
        float t  = ti + s0[e];
        float sg = __builtin_amdgcn_rcpf(1.0f + __expf(-t));  // v_exp + v_rcp
        r0[e] = (m0[e] == 0) ? MASK_FILL : fmaf(P_ASEM, sg, Q_ASEM * a0[e]);
    }
#pragma unroll
    for (int e = 0; e < 4; ++e) {
        float t  = ti + s1[e];
        float sg = __builtin_amdgcn_rcpf(1.0f + __expf(-t));
        r1[e] = (m1[e] == 0) ? MASK_FILL : fmaf(P_ASEM, sg, Q_ASEM * a1[e]);
    }

    __builtin_nontemporal_store(r0, (f4*)orow);
    __builtin_nontemporal_store(r1, (f4*)(orow + 4));
}

extern "C" void kernel_launch(void* const* d_in, const int* in_sizes, int n_in,
                              void* d_out, int out_size, void* d_ws, size_t ws_size,
                              hipStream_t stream) {
    const float* x    = (const float*)d_in[0];  // (16,2048,256) f32
    const float* adj  = (const float*)d_in[1];  // (16,2048,2048) f32
    const int*   mask = (const int*)  d_in[2];  // (16,2048) i32
    const float* W    = (const float*)d_in[3];  // (1,512) f32
    const float* bptr = (const float*)d_in[4];  // (1,) f32
    float*       out  = (float*)d_out;          // (16,2048,2048) f32

    float* si = (float*)d_ws;                   // NROWS floats
    float* sj = si + NROWS;                     // NROWS floats (256 KB total)

    // Kernel 1: 2048 16-row tiles, 8 waves (tiles) per 256-thread block.
    sisj_wmma_kernel<<<NROWS / 16 / 8, 256, 0, stream>>>(x, W, si, sj);

    // Kernel 2: one block per (b,i) row.
    fuse_kernel<<<NROWS, 256, 0, stream>>>(adj, mask, si, sj, bptr, out);
}